// CoAttention_7988639170744
// MI455X (gfx1250) — compile-verified
//
#include <hip/hip_runtime.h>
#include <hip/hip_bf16.h>

// ---------------------------------------------------------------------------
// CoAttention (hyperbolic co-attention) for MI455X / gfx1250, wave32 + WMMA.
// fp32 end-to-end; big GEMMs use V_WMMA_F32_16X16X4_F32 with fully padded
// operands (no bounds guards in the hot loop), 2x2 accumulator macro-tiles.
// ---------------------------------------------------------------------------

typedef __attribute__((ext_vector_type(2))) float v2f;
typedef __attribute__((ext_vector_type(8))) float v8f;

#define F_EPS  1e-15f
#define F_MAXN 0.996f            // 1.0 - 0.004

// ---------------- wave32 / block reduction helpers -------------------------

__device__ __forceinline__ float wave_sum_f(float v) {
#pragma unroll
  for (int o = 16; o > 0; o >>= 1) v += __shfl_xor(v, o, 32);
  return v;
}

__device__ __forceinline__ float wave_max_f(float v) {
#pragma unroll
  for (int o = 16; o > 0; o >>= 1) v = fmaxf(v, __shfl_xor(v, o, 32));
  return v;
}

__device__ __forceinline__ float block_sum_f(float v, float* sh) {
  int lane = threadIdx.x & 31, wid = threadIdx.x >> 5, nw = blockDim.x >> 5;
  v = wave_sum_f(v);
  __syncthreads();
  if (lane == 0) sh[wid] = v;
  __syncthreads();
  float r = ((int)threadIdx.x < nw) ? sh[threadIdx.x] : 0.f;
  if (wid == 0) { r = wave_sum_f(r); if (lane == 0) sh[0] = r; }
  __syncthreads();
  return sh[0];
}

__device__ __forceinline__ float block_max_f(float v, float* sh) {
  int lane = threadIdx.x & 31, wid = threadIdx.x >> 5, nw = blockDim.x >> 5;
  v = wave_max_f(v);
  __syncthreads();
  if (lane == 0) sh[wid] = v;
  __syncthreads();
  float r = ((int)threadIdx.x < nw) ? sh[threadIdx.x] : -3.0e38f;
  if (wid == 0) { r = wave_max_f(r); if (lane == 0) sh[0] = r; }
  __syncthreads();
  return sh[0];
}

__device__ __forceinline__ float artanh_clip(float x) {
  x = fminf(fmaxf(x, -1.f + 1e-5f), 1.f - 1e-5f);
  return 0.5f * logf((1.f + x) / (1.f - x));
}

// ---------------- strided batched WMMA f32 GEMM ----------------------------
// C(m,n) = sum_k A[m*a_rs + k*a_cs] * B[k*b_rs + n*b_cs]
// Requirements (guaranteed by padding at call sites):
//   M % 128 == 0, N % 32 == 0, K % 8 == 0; padded K region is zero-filled.
// Each wave: 32x32 tile = 2x2 WMMA accumulators; K stepped by 8 (2 WMMA
// k-slices per step) -> 8 v_wmma per loop body, 4 WMMAs per fragment pair.
// AC: a_cs == 1 (K contiguous in A) -> aligned v2f (b64) loads.
// BC: b_rs == 1 (K contiguous in B) -> aligned v2f (b64) loads.

template <bool AC, bool BC>
__global__ void __launch_bounds__(128)
k_wmma_gemm(const float* __restrict__ A, const float* __restrict__ B,
            float* __restrict__ C, int K,
            long a_rs, long a_cs, long a_bs,
            long b_rs, long b_cs, long b_bs,
            long c_rs, long c_bs) {
  const int wave = threadIdx.x >> 5;
  const int lane = threadIdx.x & 31;
  const long m0 = ((long)blockIdx.x * 4 + wave) * 32;
  const long n0 = (long)blockIdx.y * 32;
  const int r  = lane & 15;                 // row (A) / col (B,C) within tile
  const int kh = lane >> 4;                 // K-half select

  const float* pA0 = A + (long)blockIdx.z * a_bs + (m0 + r)      * a_rs;
  const float* pA1 = A + (long)blockIdx.z * a_bs + (m0 + 16 + r) * a_rs;
  const float* pB0 = B + (long)blockIdx.z * b_bs + (n0 + r)      * b_cs;
  const float* pB1 = B + (long)blockIdx.z * b_bs + (n0 + 16 + r) * b_cs;

  v8f acc00 = {0.f,0.f,0.f,0.f,0.f,0.f,0.f,0.f};
  v8f acc01 = acc00, acc10 = acc00, acc11 = acc00;

  for (int k0 = 0; k0 < K; k0 += 8) {
    // speculative prefetch of the next K-step (global_prefetch path)
    {
      const long kp = k0 + 8 + 2 * kh;
      __builtin_prefetch(AC ? (pA0 + kp) : (pA0 + kp * a_cs), 0, 3);
      __builtin_prefetch(BC ? (pB0 + kp) : (pB0 + kp * b_rs), 0, 3);
    }
#pragma unroll
    for (int u = 0; u < 2; ++u) {
      const long ka = (long)k0 + 4 * u + 2 * kh;
      v2f a0, a1, b0, b1;
      if (AC) {
        a0 = *(const v2f*)(pA0 + ka);
        a1 = *(const v2f*)(pA1 + ka);
      } else {
        a0.x = pA0[ka * a_cs]; a0.y = pA0[(ka + 1) * a_cs];
        a1.x = pA1[ka * a_cs]; a1.y = pA1[(ka + 1) * a_cs];
      }
      if (BC) {
        b0 = *(const v2f*)(pB0 + ka);
        b1 = *(const v2f*)(pB1 + ka);
      } else {
        b0.x = pB0[ka * b_rs]; b0.y = pB0[(ka + 1) * b_rs];
        b1.x = pB1[ka * b_rs]; b1.y = pB1[(ka + 1) * b_rs];
      }
      acc00 = __builtin_amdgcn_wmma_f32_16x16x4_f32(false, a0, false, b0, (short)0, acc00, false, false);
      acc01 = __builtin_amdgcn_wmma_f32_16x16x4_f32(false, a0, false, b1, (short)0, acc01, false, false);
      acc10 = __builtin_amdgcn_wmma_f32_16x16x4_f32(false, a1, false, b0, (short)0, acc10, false, false);
      acc11 = __builtin_amdgcn_wmma_f32_16x16x4_f32(false, a1, false, b1, (short)0, acc11, false, false);
    }
  }

  float* pC = C + (long)blockIdx.z * c_bs;
#pragma unroll
  for (int i = 0; i < 8; ++i) {
    const long mc0 = m0 + kh * 8 + i;
    const long mc1 = mc0 + 16;
    pC[mc0 * c_rs + n0 + r]      = acc00[i];
    pC[mc0 * c_rs + n0 + 16 + r] = acc01[i];
    pC[mc1 * c_rs + n0 + r]      = acc10[i];
    pC[mc1 * c_rs + n0 + 16 + r] = acc11[i];
  }
}

// ---------------- weight pack: W (N x 101) -> zero-padded (Npad x 104) -----

__global__ void k_packw(const float* __restrict__ W, float* __restrict__ Wp,
                        int N, int Npad) {
  const int idx = blockIdx.x * blockDim.x + threadIdx.x;   // n*104 + k
  if (idx >= Npad * 104) return;
  const int n = idx / 104, k = idx % 104;
  Wp[idx] = (n < N && k < 101) ? W[n * 101 + k] : 0.f;
}

// ---------------- poincare_to_lorentz: (rows,100) -> (rows,104 pad) --------

__global__ void k_p2l(const float* __restrict__ x, float* __restrict__ out, int rows) {
  const int row = blockIdx.x * (blockDim.x >> 5) + (threadIdx.x >> 5);
  if (row >= rows) return;
  const int lane = threadIdx.x & 31;
  const float* xr = x + (long)row * 100;
  float v[4]; float s = 0.f;
#pragma unroll
  for (int t = 0; t < 4; ++t) {
    const int j = lane + 32 * t;
    v[t] = (j < 100) ? xr[j] : 0.f;
    s += v[t] * v[t];
  }
  const float x2 = wave_sum_f(s);
  const float den = 1.f - x2 + 1e-6f;
  float* o = out + (long)row * 104;
  if (lane == 0) o[0] = (1.f + x2) / den;
#pragma unroll
  for (int t = 0; t < 4; ++t) {
    const int j = lane + 32 * t;
    if (j < 100) o[1 + j] = 2.f * v[t] / den;
  }
  if (lane < 3) o[101 + lane] = 0.f;        // zero K-padding
}

// ---------------- l_linear post (bias + hyperboloid lift) ------------------
// H stride 160.  to_poincare: out stride 128 (drop time, /(t+1)).
// else: out stride 104 lorentz, zero-padded cols 101..103.

__global__ void k_linpost(const float* __restrict__ H, const float* __restrict__ bias,
                          const float* __restrict__ sp, float* __restrict__ out,
                          int rows, int Dout, int to_poincare) {
  const int row = blockIdx.x * (blockDim.x >> 5) + (threadIdx.x >> 5);
  if (row >= rows) return;
  const int lane = threadIdx.x & 31;
  const float* hr = H + (long)row * 160;
  const float s = sp[0];
  const float h0 = hr[0] + bias[0];
  const float tval = (1.f / (1.f + expf(-h0))) * s + 1.1f;
  float v[4]; float sv = 0.f;
#pragma unroll
  for (int t = 0; t < 4; ++t) {
    const int j = 1 + lane + 32 * t;
    v[t] = (j < Dout) ? (hr[j] + bias[j]) : 0.f;
    sv += v[t] * v[t];
  }
  sv = wave_sum_f(sv);
  const float sc = sqrtf((tval * tval - 1.f) / fmaxf(sv, 1e-8f));
  if (to_poincare) {
    const float inv = 1.f / (tval + 1.f);
#pragma unroll
    for (int t = 0; t < 4; ++t) {
      const int j = 1 + lane + 32 * t;
      if (j < Dout) out[(long)row * 128 + (j - 1)] = v[t] * sc * inv;
    }
  } else {
    if (lane == 0) out[(long)row * 104] = tval;
#pragma unroll
    for (int t = 0; t < 4; ++t) {
      const int j = 1 + lane + 32 * t;
      if (j < Dout) out[(long)row * 104 + j] = v[t] * sc;
    }
    if (lane < 3) out[(long)row * 104 + 101 + lane] = 0.f;   // zero K-padding
  }
}

// ---------------- l_act on rows of S (len 1024), in-place; stores row norm -

__global__ void __launch_bounds__(256)
k_lact(float* __restrict__ S, float* __restrict__ rownorm) {
  const long row = blockIdx.x;
  float* xr = S + row * 1024;
  const int tid = threadIdx.x;
  __shared__ float sh[8];
  __shared__ float s0;
  float x[4];
#pragma unroll
  for (int t = 0; t < 4; ++t) x[t] = xr[tid + 256 * t];
  if (tid == 0) s0 = x[0];
  float sum1 = 0.f;
#pragma unroll
  for (int t = 0; t < 4; ++t) {
    const int j = tid + 256 * t;
    if (j >= 1) sum1 += x[t] * x[t];
  }
  const float n = fmaxf(sqrtf(block_sum_f(sum1, sh)), 1e-8f);   // also syncs s0
  const float tt = fmaxf(s0, 1.f + 1e-7f);
  const float d = acoshf(tt);
  float w[4]; float sum2 = 0.f;
#pragma unroll
  for (int t = 0; t < 4; ++t) {
    const int j = tid + 256 * t;
    w[t] = (j >= 1) ? tanhf(d * x[t] / n) : 0.f;
    sum2 += w[t] * w[t];
  }
  const float nn = fmaxf(sqrtf(block_sum_f(sum2, sh)), 1e-8f);
  const float ch = coshf(nn);
  const float sf = sinhf(nn) / nn;
#pragma unroll
  for (int t = 0; t < 4; ++t) {
    const int j = tid + 256 * t;
    xr[j] = (j == 0) ? ch : sf * w[t];
  }
  if (tid == 0) {
    const float snh = sinhf(nn);
    rownorm[row] = sqrtf(ch * ch + snh * snh);
  }
}

// ---------------- column norms of S: cn[b*1024+j] = ||S[b,:,j]|| -----------

__global__ void k_colnorm(const float* __restrict__ S, float* __restrict__ cn) {
  const long idx = (long)blockIdx.x * blockDim.x + threadIdx.x;   // b*1024+j
  const long b = idx >> 10, j = idx & 1023;
  const float* base = S + (b << 20) + j;
  float s = 0.f;
  for (int i = 0; i < 1024; ++i) { const float v = base[(long)i * 1024]; s += v * v; }
  cn[idx] = sqrtf(s);
}

// ---------------- mobius_matvec normalization (rows of 128, in-place) ------

__global__ void k_mmv_post(float* __restrict__ MX, const float* __restrict__ xn_arr,
                           int rows) {
  const int row = blockIdx.x * (blockDim.x >> 5) + (threadIdx.x >> 5);
  if (row >= rows) return;
  const int lane = threadIdx.x & 31;
  float* r = MX + (long)row * 128;
  float m[4]; float s2 = 0.f;
#pragma unroll
  for (int t = 0; t < 4; ++t) { m[t] = r[lane + 32 * t]; s2 += m[t] * m[t]; }
  s2 = wave_sum_f(s2);
  const float mxn = fmaxf(sqrtf(s2), F_EPS);
  const float xn = fmaxf(xn_arr[row], F_EPS);
  const float tv = tanhf(mxn / xn * artanh_clip(xn));
  float f = (s2 == 0.f) ? 0.f : tv / mxn;
  const float nrm = fabsf(tv);
  if (nrm > F_MAXN) f *= F_MAXN / nrm;
#pragma unroll
  for (int t = 0; t < 4; ++t) r[lane + 32 * t] = f * m[t];
}

// ---------------- Hs: mobius_add + p_expmap0(tanh(p_logmap0)) (128-dim) ----

__global__ void k_hs_combine(const float* __restrict__ XA, float* __restrict__ YB,
                             int rows) {
  const int row = blockIdx.x * (blockDim.x >> 5) + (threadIdx.x >> 5);
  if (row >= rows) return;
  const int lane = threadIdx.x & 31;
  const float* xr = XA + (long)row * 128;
  float* yr = YB + (long)row * 128;
  float x[4], y[4];
  float sx = 0.f, sy = 0.f, sxy = 0.f;
#pragma unroll
  for (int t = 0; t < 4; ++t) {
    x[t] = xr[lane + 32 * t]; y[t] = yr[lane + 32 * t];
    sx += x[t] * x[t]; sy += y[t] * y[t]; sxy += x[t] * y[t];
  }
  const float x2 = wave_sum_f(sx);
  const float y2 = wave_sum_f(sy);
  const float xy = wave_sum_f(sxy);
  const float ca = 1.f + 2.f * xy + y2;
  const float cb = 1.f - x2;
  const float den = fmaxf(1.f + 2.f * xy + x2 * y2, F_EPS);
  float z[4]; float sz = 0.f;
#pragma unroll
  for (int t = 0; t < 4; ++t) { z[t] = (ca * x[t] + cb * y[t]) / den; sz += z[t] * z[t]; }
  const float nz = sqrtf(wave_sum_f(sz));
  const float f1 = (nz > F_MAXN) ? (F_MAXN / nz) : 1.f;      // proj_ball
  const float n1 = fmaxf(fminf(nz, F_MAXN), F_EPS);
  const float g = artanh_clip(n1) / n1 * f1;                 // p_logmap0 scale
  float w[4]; float sw = 0.f;
#pragma unroll
  for (int t = 0; t < 4; ++t) { w[t] = tanhf(g * z[t]); sw += w[t] * w[t]; }
  const float nw = sqrtf(wave_sum_f(sw));
  const float n2 = fmaxf(nw, F_EPS);
  const float t2 = tanhf(n2);
  const float on = t2 * nw / n2;
  const float f2 = (t2 / n2) * ((on > F_MAXN) ? (F_MAXN / on) : 1.f);
#pragma unroll
  for (int t = 0; t < 4; ++t) yr[lane + 32 * t] = f2 * w[t];
}

// ---------------- Hc (swapped): 1024-dim mobius_add + p_exp(tanh(p_log)) ---
// block per (b,k); x_j = HCA[b,j,k], y_j = HCB[b,j,k]; out HC[b,k,j].

__global__ void __launch_bounds__(256)
k_hc_combine(const float* __restrict__ HCA, const float* __restrict__ HCB,
             float* __restrict__ HC) {
  const int bk = blockIdx.x;                 // b*128 + k
  const long b = bk >> 7;
  const long k = bk & 127;
  const int tid = threadIdx.x;
  __shared__ float sh[8];
  float x[4], y[4];
  float sx = 0.f, sy = 0.f, sxy = 0.f;
#pragma unroll
  for (int t = 0; t < 4; ++t) {
    const long j = tid + 256 * t;
    const long idx = (b * 1024 + j) * 128 + k;
    x[t] = HCA[idx]; y[t] = HCB[idx];
    sx += x[t] * x[t]; sy += y[t] * y[t]; sxy += x[t] * y[t];
  }
  const float x2 = block_sum_f(sx, sh);
  const float y2 = block_sum_f(sy, sh);
  const float xy = block_sum_f(sxy, sh);
  const float ca = 1.f + 2.f * xy + y2;
  const float cb = 1.f - x2;
  const float den = fmaxf(1.f + 2.f * xy + x2 * y2, F_EPS);
  float z[4]; float sz = 0.f;
#pragma unroll
  for (int t = 0; t < 4; ++t) { z[t] = (ca * x[t] + cb * y[t]) / den; sz += z[t] * z[t]; }
  const float nz = sqrtf(block_sum_f(sz, sh));
  const float f1 = (nz > F_MAXN) ? (F_MAXN / nz) : 1.f;
  const float n1 = fmaxf(fminf(nz, F_MAXN), F_EPS);
  const float g = artanh_clip(n1) / n1 * f1;
  float w[4]; float sw = 0.f;
#pragma unroll
  for (int t = 0; t < 4; ++t) { w[t] = tanhf(g * z[t]); sw += w[t] * w[t]; }
  const float nw = sqrtf(block_sum_f(sw, sh));
  const float n2 = fmaxf(nw, F_EPS);
  const float t2 = tanhf(n2);
  const float on = t2 * nw / n2;
  const float f2 = (t2 / n2) * ((on > F_MAXN) ? (F_MAXN / on) : 1.f);
#pragma unroll
  for (int t = 0; t < 4; ++t) HC[(long)bk * 1024 + tid + 256 * t] = f2 * w[t];
}

// ---------------- mobius_matvec with 1xK weight -> scalar ------------------

__device__ __forceinline__ float mmv_scalar(float dot, float s2) {
  if (dot == 0.f) return 0.f;
  const float xn = fmaxf(sqrtf(s2), F_EPS);
  const float mxn = fmaxf(fabsf(dot), F_EPS);
  const float tv = tanhf(mxn / xn * artanh_clip(xn));
  float val = tv * (dot / mxn);
  const float n = fmaxf(fabsf(val), F_EPS);
  if (n > F_MAXN) val = val / n * F_MAXN;
  return val;
}

__global__ void k_as_row(const float* __restrict__ HS, const float* __restrict__ wh,
                         float* __restrict__ out, int rows) {
  const int row = blockIdx.x * (blockDim.x >> 5) + (threadIdx.x >> 5);
  if (row >= rows) return;
  const int lane = threadIdx.x & 31;
  const float* r = HS + (long)row * 128;
  float dot = 0.f, s2 = 0.f;
#pragma unroll
  for (int t = 0; t < 4; ++t) {
    const int j = lane + 32 * t;
    const float v = r[j];
    dot += v * wh[j]; s2 += v * v;
  }
  dot = wave_sum_f(dot);
  s2 = wave_sum_f(s2);
  if (lane == 0) out[row] = mmv_scalar(dot, s2);
}

__global__ void k_ac_row(const float* __restrict__ HC, const float* __restrict__ wh,
                         float* __restrict__ out, int rows) {
  const int row = blockIdx.x * (blockDim.x >> 5) + (threadIdx.x >> 5);
  if (row >= rows) return;
  const int lane = threadIdx.x & 31;
  const long b = row >> 10, j = row & 1023;
  const float* base = HC + b * 131072 + j;          // HC[b, k, j], k strided
  float dot = 0.f, s2 = 0.f;
#pragma unroll
  for (int t = 0; t < 4; ++t) {
    const int kk = lane + 32 * t;
    const float v = base[(long)kk * 1024];
    dot += v * wh[kk]; s2 += v * v;
  }
  dot = wave_sum_f(dot);
  s2 = wave_sum_f(s2);
  if (lane == 0) out[row] = mmv_scalar(dot, s2);
}

// ---------------- softmax over rows of 1024 --------------------------------

__global__ void __launch_bounds__(256)
k_softmax1024(const float* __restrict__ in, float* __restrict__ out) {
  const int b = blockIdx.x;
  const int tid = threadIdx.x;
  __shared__ float sh[8];
  const float* r = in + (long)b * 1024;
  float v[4]; float mx = -3.0e38f;
#pragma unroll
  for (int t = 0; t < 4; ++t) { v[t] = r[tid + 256 * t]; mx = fmaxf(mx, v[t]); }
  mx = block_max_f(mx, sh);
  float s = 0.f;
#pragma unroll
  for (int t = 0; t < 4; ++t) { v[t] = expf(v[t] - mx); s += v[t]; }
  s = block_sum_f(s, sh);
  const float inv = 1.f / s;
#pragma unroll
  for (int t = 0; t < 4; ++t) out[(long)b * 1024 + tid + 256 * t] = v[t] * inv;
}

// ---------------- l_centroid: avg = w @ Lz, normalize by Lorentz norm ------

__global__ void __launch_bounds__(256)
k_centroid(const float* __restrict__ Lz, const float* __restrict__ w,
           float* __restrict__ out) {
  const int b = blockIdx.x;
  const int d = threadIdx.x;
  __shared__ float sh[8];
  __shared__ float a0s;
  float acc = 0.f;
  if (d < 101) {
    const float* wr = w + (long)b * 1024;
    const float* lr = Lz + (long)b * 1024 * 104 + d;
    for (int i = 0; i < 1024; ++i) acc += wr[i] * lr[(long)i * 104];
  }
  if (d == 0) a0s = acc;
  const float contrib = (d >= 1 && d < 101) ? acc * acc : 0.f;
  const float ssum = block_sum_f(contrib, sh);     // syncs a0s too
  const float inner = ssum - a0s * a0s;
  const float den = sqrtf(fmaxf(fabsf(inner), 1e-8f));
  if (d < 101) out[(long)b * 104 + d] = acc / den;
}

// ---------------- l_concat -> co_sc (32 x 201) -----------------------------

__global__ void __launch_bounds__(256)
k_concat(const float* __restrict__ cs, const float* __restrict__ cc,
         float* __restrict__ out) {
  const int b = blockIdx.x, tid = threadIdx.x;
  __shared__ float sh[8];
  float s = 0.f;
  if (tid < 200)
    s = (tid < 100) ? cs[(long)b * 104 + 1 + tid] : cc[(long)b * 104 + 1 + (tid - 100)];
  const float ss = block_sum_f(s * s, sh);
  const float t = sqrtf(1.f + ss);
  if (tid == 0) out[(long)b * 201] = t;
  if (tid < 200) out[(long)b * 201 + 1 + tid] = s;
}

// ---------------------------------------------------------------------------

extern "C" void kernel_launch(void* const* d_in, const int* in_sizes, int n_in,
                              void* d_out, int out_size, void* d_ws, size_t ws_size,
                              hipStream_t stream) {
  (void)in_sizes; (void)n_in; (void)out_size; (void)ws_size;
  const float* sent = (const float*)d_in[0];
  const float* comm = (const float*)d_in[1];
  const float* Wl_w = (const float*)d_in[2];
  const float* Wl_b = (const float*)d_in[3];
  const float* Wl_s = (const float*)d_in[4];
  const float* Ws_w = (const float*)d_in[5];
  const float* Ws_b = (const float*)d_in[6];
  const float* Ws_s = (const float*)d_in[7];
  const float* Wc_w = (const float*)d_in[8];
  const float* Wc_b = (const float*)d_in[9];
  const float* Wc_s = (const float*)d_in[10];
  const float* whs  = (const float*)d_in[11];
  const float* whc  = (const float*)d_in[12];

  const int ROWS = 32 * 1024;                // 32768
  float* ws = (float*)d_ws;
  long off = 0;
  float* LS  = ws + off; off += (long)ROWS * 104;
  float* LC  = ws + off; off += (long)ROWS * 104;
  float* LM  = ws + off; off += (long)ROWS * 104;
  float* HH  = ws + off; off += (long)ROWS * 160;   // l_linear GEMM scratch
  float* S   = ws + off; off += (long)32 * 1024 * 1024;
  float* RN  = ws + off; off += ROWS;               // row norms of S
  float* CN  = ws + off; off += ROWS;               // col norms of S
  float* HSA = ws + off; off += (long)ROWS * 128;
  float* HCA = ws + off; off += (long)ROWS * 128;
  float* HSB = ws + off; off += (long)ROWS * 128;   // becomes Hs
  float* HCB = ws + off; off += (long)ROWS * 128;
  float* HC  = ws + off; off += (long)ROWS * 128;   // (32,128,1024)
  float* ASR = ws + off; off += 32 * 1024;
  float* ACR = ws + off; off += 32 * 1024;
  float* CS  = ws + off; off += 32 * 104;
  float* CC  = ws + off; off += 32 * 104;
  float* WPL = ws + off; off += 128 * 104;          // padded Wl (128 x 104)
  float* WPS = ws + off; off += 160 * 104;          // padded Ws (160 x 104)
  float* WPC = ws + off; off += 160 * 104;          // padded Wc (160 x 104)

  float* out_cosc = (float*)d_out;                  // 32*201
  float* out_as   = out_cosc + 32 * 201;            // 32*1024
  float* out_ac   = out_as + 32 * 1024;             // 32*1024

  // 0) pack zero-padded weights
  k_packw<<<(128 * 104 + 255) / 256, 256, 0, stream>>>(Wl_w, WPL, 101, 128);
  k_packw<<<(160 * 104 + 255) / 256, 256, 0, stream>>>(Ws_w, WPS, 129, 160);
  k_packw<<<(160 * 104 + 255) / 256, 256, 0, stream>>>(Wc_w, WPC, 129, 160);

  // 1) Lorentz lifts (zero K-padding written)
  k_p2l<<<ROWS / 8, 256, 0, stream>>>(sent, LS, ROWS);
  k_p2l<<<ROWS / 8, 256, 0, stream>>>(comm, LC, ROWS);

  // 2) L = l_linear(lc, Wl)  (lorentz output, 101 dims, stride 104)
  {
    dim3 g(ROWS / 128, 128 / 32, 1);
    k_wmma_gemm<true, true><<<g, 128, 0, stream>>>(
        LC, WPL, HH, 104, 104, 1, 0, 1, 104, 0, 160, 0);
  }
  k_linpost<<<ROWS / 8, 256, 0, stream>>>(HH, Wl_b, Wl_s, LM, ROWS, 101, 0);

  // 3) S = ls @ L^T  (batched over 32; K padded to 104 with zeros)
  {
    dim3 g(1024 / 128, 1024 / 32, 32);
    k_wmma_gemm<true, true><<<g, 128, 0, stream>>>(
        LS, LM, S, 104,
        104, 1, (long)1024 * 104, 1, 104, (long)1024 * 104,
        1024, (long)1024 * 1024);
  }

  // 4) S = l_act(S) (in place) + row norms; then column norms
  k_lact<<<ROWS, 256, 0, stream>>>(S, RN);
  k_colnorm<<<ROWS / 256, 256, 0, stream>>>(S, CN);

  // 5) Hs_a = poincare(l_linear(ls, Ws)), Hc_a = poincare(l_linear(lc, Wc))
  {
    dim3 g(ROWS / 128, 160 / 32, 1);
    k_wmma_gemm<true, true><<<g, 128, 0, stream>>>(
        LS, WPS, HH, 104, 104, 1, 0, 1, 104, 0, 160, 0);
  }
  k_linpost<<<ROWS / 8, 256, 0, stream>>>(HH, Ws_b, Ws_s, HSA, ROWS, 129, 1);
  {
    dim3 g(ROWS / 128, 160 / 32, 1);
    k_wmma_gemm<true, true><<<g, 128, 0, stream>>>(
        LC, WPC, HH, 104, 104, 1, 0, 1, 104, 0, 160, 0);
  }
  k_linpost<<<ROWS / 8, 256, 0, stream>>>(HH, Wc_b, Wc_s, HCA, ROWS, 129, 1);

  // 6) Hs_b raw = S @ Hc_a, then mobius_matvec normalization (row norms)
  {
    dim3 g(1024 / 128, 128 / 32, 32);
    k_wmma_gemm<true, false><<<g, 128, 0, stream>>>(
        S, HCA, HSB, 1024,
        1024, 1, (long)1024 * 1024, 128, 1, (long)1024 * 128,
        128, (long)1024 * 128);
  }
  k_mmv_post<<<ROWS / 8, 256, 0, stream>>>(HSB, RN, ROWS);

  // 7) Hc_b raw = S^T @ Hs_a, normalize with column norms
  {
    dim3 g(1024 / 128, 128 / 32, 32);
    k_wmma_gemm<false, false><<<g, 128, 0, stream>>>(
        S, HSA, HCB, 1024,
        1, 1024, (long)1024 * 1024, 128, 1, (long)1024 * 128,
        128, (long)1024 * 128);
  }
  k_mmv_post<<<ROWS / 8, 256, 0, stream>>>(HCB, CN, ROWS);

  // 8) Hs = p_exp(tanh(p_log(mobius_add(Hs_a, Hs_b))))  [in place -> HSB]
  k_hs_combine<<<ROWS / 8, 256, 0, stream>>>(HSA, HSB, ROWS);

  // 9) Hc (swapped, 1024-dim rows)
  k_hc_combine<<<32 * 128, 256, 0, stream>>>(HCA, HCB, HC);

  // 10) attention logits + softmax
  k_as_row<<<ROWS / 8, 256, 0, stream>>>(HSB, whs, ASR, ROWS);
  k_ac_row<<<ROWS / 8, 256, 0, stream>>>(HC, whc, ACR, ROWS);
  k_softmax1024<<<32, 256, 0, stream>>>(ASR, out_as);
  k_softmax1024<<<32, 256, 0, stream>>>(ACR, out_ac);

  // 11) centroids + concat
  k_centroid<<<32, 256, 0, stream>>>(LS, out_as, CS);
  k_centroid<<<32, 256, 0, stream>>>(LC, out_ac, CC);
  k_concat<<<32, 256, 0, stream>>>(CS, CC, out_cosc);
}